// UniPhyFluidScan_88098369176121
// MI455X (gfx1250) — compile-verified
//
#include <hip/hip_runtime.h>
#include <hip/hip_bf16.h>
#include <math.h>

// UniPhyFluidScan for MI455X (gfx1250, wave32).
// DFT stages implemented as dense GEMMs on V_WMMA_F32_16X16X4_F32 (full fp32 WMMA).
// Streaming strided loops issue global_prefetch_b8 (speculative) one plane ahead.

#define B_ 4
#define T_ 32
#define C_ 8
#define C2_ 16
#define H_ 128
#define W_ 128
#define WF_ 65
#define WFP_ 80              // Wf padded to a multiple of 16 (zeroed twiddle cols)
#define BT_ 128              // B*T
#define HW_ 16384            // H*W
#define ROWS_ 131072         // BT*C*H  (rows for row-direction DFTs)
#define NPLANE_ ((size_t)BT_*C_*H_*WFP_)   // 10,485,760 floats per complex plane

typedef float v2f __attribute__((ext_vector_type(2)));
typedef float v8f __attribute__((ext_vector_type(8)));

__device__ __forceinline__ float siluf(float z){ float s=1.f/(1.f+__expf(-z)); return z*s; }
__device__ __forceinline__ float dsiluf(float z){ float s=1.f/(1.f+__expf(-z)); return s*(1.f+z*(1.f-s)); }

// D = A(16x4,f32) * B(4x16,f32) + C(16x16,f32), wave32.
// A layout (ISA 7.12.2): lane l: m=l&15, VGPR0/1 = K = 2*(l>>4), 2*(l>>4)+1.
// B layout assumed to mirror A (M<->N): lane l: n=l&15, VGPR0/1 = K = 2*(l>>4), +1.
// C/D: VGPR j: lanes0-15 -> M=j, lanes16-31 -> M=j+8; N = lane&15.
__device__ __forceinline__ v8f wmma4(v2f a, v2f b, v8f c){
  return __builtin_amdgcn_wmma_f32_16x16x4_f32(false, a, false, b, (short)0, c, false, false);
}

// ---------------------------------------------------------------- tables ----
__global__ void k_init(float* twc, float* tws, float* thc, float* ths,
                       float* tiwc, float* tiws, float* kg, float* nin, float* nout)
{
  int i = blockIdx.x*blockDim.x + threadIdx.x;   // 16384 threads cover all tables
  const float PI2 = 6.283185307179586f;
  if (i < H_*WFP_) {                 // forward row-DFT: TW[w][k] = e^{-2pi i wk/128}
    int w = i / WFP_, k = i % WFP_;
    float c=0.f, s=0.f;
    if (k < WF_) { float ang = -PI2*(float)(w*k)/128.f; c = cosf(ang); s = sinf(ang); }
    twc[i]=c; tws[i]=s;              // cols >=65 zero -> padded outputs exactly 0
  }
  if (i < H_*H_) {                   // col-DFT: TH[kh][h]: cos/sin(2pi kh h/128)
    int kh = i / H_, h = i % H_;
    float ang = PI2*(float)(kh*h)/128.f;
    thc[i]=cosf(ang); ths[i]=sinf(ang);
  }
  if (i < WFP_*W_) {                 // inverse row-DFT (real out, hermitian weights)
    int k = i / W_, w = i % W_;
    float c=0.f, s=0.f;
    if (k < WF_) {
      float wt = (k==0 || k==64) ? 1.f : 2.f;
      float ang = PI2*(float)(k*w)/128.f;
      c = wt*cosf(ang); s = -wt*sinf(ang);
    }
    tiwc[i]=c; tiws[i]=s;            // rows >=65 zero
  }
  if (i < H_*WFP_) {                 // k-grid: (gky, gkx, k2); pads are identity (0,0,1)
    int h = i / WFP_, k = i % WFP_;
    float gy=0.f, gx=0.f, k2=1.f;
    if (k < WF_) {
      gy = (h < 64 ? (float)h : (float)(h-128)) / 128.f;
      gx = (float)k / 128.f;
      k2 = gy*gy + gx*gx;
      if (h==0 && k==0) k2 = 1.f;
    }
    kg[3*i]=gy; kg[3*i+1]=gx; kg[3*i+2]=k2;
  }
  if (i < BT_) { nin[i]=0.f; nout[i]=0.f; }
}

// ------------------------------------------------------------- propagator ---
__global__ void k_A(const float* pw1, const float* pb1, const float* pw2, const float* pb2,
                    const float* dt, const float* kg, float* Ar, float* Ai)
{
  int i = blockIdx.x*blockDim.x + threadIdx.x;   // over H*WFP
  if (i >= H_*WFP_) return;
  float gy = kg[3*i], gx = kg[3*i+1];
  float lrn[C_];
  #pragma unroll
  for (int c=0;c<C_;c++) lrn[c]=pb2[c];
  for (int j=0;j<64;j++){
    float hid = siluf(gy*pw1[j] + gx*pw1[64+j] + pb1[j]);
    #pragma unroll
    for (int c=0;c<C_;c++) lrn[c] += hid*pw2[j*C_+c];
  }
  float kp = sqrtf(gy*gy+gx*gx);
  for (int b=0;b<B_;b++){
    float d = dt[b];
    #pragma unroll
    for (int c=0;c<C_;c++){
      float s, co; sincosf((lrn[c]+kp)*d, &s, &co);
      size_t o = (size_t)(b*C_+c)*(H_*WFP_) + i;
      Ar[o]=co; Ai[o]=s;
    }
  }
}

// ------------------------------------------------------- interaction convs --
__global__ void k_conv1(const float* x, const float* w1, const float* b1, float* z1)
{
  __shared__ float lw[C2_*C_*9];
  __shared__ float lb[C2_];
  int tid = threadIdx.x;
  for (int i=tid;i<C2_*C_*9;i+=blockDim.x) lw[i]=w1[i];
  if (tid<C2_) lb[tid]=b1[tid];
  __syncthreads();
  int pix = blockIdx.x*blockDim.x + tid;
  if (pix >= BT_*HW_) return;
  int n = pix / HW_, hw = pix % HW_, h = hw / W_, w = hw % W_;
  float acc[C2_];
  #pragma unroll
  for (int o=0;o<C2_;o++) acc[o]=lb[o];
  const float* xin = x + (size_t)n*C_*HW_;
  for (int c=0;c<C_;c++){
    if (c+1 < C_) __builtin_prefetch(xin + (c+1)*HW_ + hw, 0, 1);  // next plane
    for (int ky=0;ky<3;ky++){
      int yy=h+ky-1; if ((unsigned)yy>=(unsigned)H_) continue;
      for (int kx=0;kx<3;kx++){
        int xx=w+kx-1; if ((unsigned)xx>=(unsigned)W_) continue;
        float v = xin[c*HW_ + yy*W_ + xx];
        #pragma unroll
        for (int o=0;o<C2_;o++) acc[o] += lw[(o*C_+c)*9 + ky*3 + kx]*v;
      }
    }
  }
  float* zo = z1 + (size_t)n*C2_*HW_ + hw;
  #pragma unroll
  for (int o=0;o<C2_;o++) zo[o*HW_] = acc[o];
}

__global__ void k_conv2(const float* z1, const float* w2, const float* b2, float* z2)
{
  __shared__ float lw[C2_*C2_*9];
  __shared__ float lb[C2_];
  int tid = threadIdx.x;
  for (int i=tid;i<C2_*C2_*9;i+=blockDim.x) lw[i]=w2[i];
  if (tid<C2_) lb[tid]=b2[tid];
  __syncthreads();
  int pix = blockIdx.x*blockDim.x + tid;
  if (pix >= BT_*HW_) return;
  int n = pix / HW_, hw = pix % HW_, h = hw / W_, w = hw % W_;
  float acc[C2_];
  #pragma unroll
  for (int o=0;o<C2_;o++) acc[o]=lb[o];
  const float* zin = z1 + (size_t)n*C2_*HW_;
  for (int c=0;c<C2_;c++){
    if (c+1 < C2_) __builtin_prefetch(zin + (c+1)*HW_ + hw, 0, 1);
    for (int ky=0;ky<3;ky++){
      int yy=h+ky-1; if ((unsigned)yy>=(unsigned)H_) continue;
      for (int kx=0;kx<3;kx++){
        int xx=w+kx-1; if ((unsigned)xx>=(unsigned)W_) continue;
        float v = siluf(zin[c*HW_ + yy*W_ + xx]);
        #pragma unroll
        for (int o=0;o<C2_;o++) acc[o] += lw[(o*C2_+c)*9 + ky*3 + kx]*v;
      }
    }
  }
  float* zo = z2 + (size_t)n*C2_*HW_ + hw;
  #pragma unroll
  for (int o=0;o<C2_;o++) zo[o*HW_] = acc[o];
}

// grad wrt conv3 input is per-channel constant gbar2[c] = sum_o w3[o][c]
__global__ void k_g2(float* z2, const float* w3)
{
  int i = blockIdx.x*blockDim.x + threadIdx.x;
  if (i >= BT_*C2_*HW_) return;
  int c = (i / HW_) % C2_;
  float gbar = 0.f;
  #pragma unroll
  for (int c3=0;c3<C_;c3++) gbar += w3[c3*C2_ + c];
  z2[i] = gbar * dsiluf(z2[i]);
}

// g1 = silu'(z1) * conv2^T(g2); writes over z1 (only same-pixel z1 reads -> safe)
__global__ void k_backT2(float* z1, const float* g2all, const float* w2)
{
  __shared__ float lw[C2_*C2_*9];
  int tid = threadIdx.x;
  for (int i=tid;i<C2_*C2_*9;i+=blockDim.x) lw[i]=w2[i];
  __syncthreads();
  int pix = blockIdx.x*blockDim.x + tid;
  if (pix >= BT_*HW_) return;
  int n = pix / HW_, hw = pix % HW_, h = hw / W_, w = hw % W_;
  float acc[C2_];
  #pragma unroll
  for (int c=0;c<C2_;c++) acc[c]=0.f;
  const float* g2 = g2all + (size_t)n*C2_*HW_;
  for (int o=0;o<C2_;o++){
    if (o+1 < C2_) __builtin_prefetch(g2 + (o+1)*HW_ + hw, 0, 1);
    for (int ky=0;ky<3;ky++){
      int yy=h+1-ky; if ((unsigned)yy>=(unsigned)H_) continue;
      for (int kx=0;kx<3;kx++){
        int xx=w+1-kx; if ((unsigned)xx>=(unsigned)W_) continue;
        float gv = g2[o*HW_ + yy*W_ + xx];
        #pragma unroll
        for (int c=0;c<C2_;c++) acc[c] += lw[(o*C2_+c)*9 + ky*3 + kx]*gv;
      }
    }
  }
  float* zp = z1 + (size_t)n*C2_*HW_ + hw;
  #pragma unroll
  for (int c=0;c<C2_;c++) zp[c*HW_] = dsiluf(zp[c*HW_]) * acc[c];
}

// X_total = x + dt[b] * conv1^T(g1)
__global__ void k_forceX(const float* x, const float* g1all, const float* w1,
                         const float* dt, float* xt)
{
  __shared__ float lw[C2_*C_*9];
  int tid = threadIdx.x;
  for (int i=tid;i<C2_*C_*9;i+=blockDim.x) lw[i]=w1[i];
  __syncthreads();
  int pix = blockIdx.x*blockDim.x + tid;
  if (pix >= BT_*HW_) return;
  int n = pix / HW_, hw = pix % HW_, h = hw / W_, w = hw % W_;
  float acc[C_];
  #pragma unroll
  for (int c=0;c<C_;c++) acc[c]=0.f;
  const float* g = g1all + (size_t)n*C2_*HW_;
  for (int o=0;o<C2_;o++){
    if (o+1 < C2_) __builtin_prefetch(g + (o+1)*HW_ + hw, 0, 1);
    for (int ky=0;ky<3;ky++){
      int yy=h+1-ky; if ((unsigned)yy>=(unsigned)H_) continue;
      for (int kx=0;kx<3;kx++){
        int xx=w+1-kx; if ((unsigned)xx>=(unsigned)W_) continue;
        float gv = g[o*HW_ + yy*W_ + xx];
        #pragma unroll
        for (int c=0;c<C_;c++) acc[c] += lw[(o*C_+c)*9 + ky*3 + kx]*gv;
      }
    }
  }
  float d = dt[n / T_];
  const float* xin = x + (size_t)n*C_*HW_ + hw;
  float* xo = xt + (size_t)n*C_*HW_ + hw;
  #pragma unroll
  for (int c=0;c<C_;c++) xo[c*HW_] = xin[c*HW_] + d*acc[c];
}

// --------------------------------------------------- WMMA DFT stages --------
// Row DFT (real in, complex out): out[r,k] = sum_w x[r,w] * TW[w][k]
__global__ void __launch_bounds__(32) k_dft_rows(const float* xt,
                                                 const float* twc, const float* tws,
                                                 float* p0, float* p1)
{
  int lane = threadIdx.x, m = lane & 15, kh2 = lane >> 4;
  int rowBase = blockIdx.x * 16, nBase = blockIdx.y * 16;
  const float* xr = xt + (size_t)rowBase*W_;
  v8f aR = {}; v8f aI = {};
  for (int k0=0;k0<W_;k0+=4){
    int ka = k0 + 2*kh2;
    v2f av = *(const v2f*)(xr + m*W_ + ka);
    v2f bc, bs;
    bc.x = twc[ka*WFP_ + nBase + m];  bc.y = twc[(ka+1)*WFP_ + nBase + m];
    bs.x = tws[ka*WFP_ + nBase + m];  bs.y = tws[(ka+1)*WFP_ + nBase + m];
    aR = wmma4(av, bc, aR);
    aI = wmma4(av, bs, aI);
  }
  #pragma unroll
  for (int j=0;j<8;j++){
    size_t o = (size_t)(rowBase + j + 8*kh2)*WFP_ + nBase + m;
    p0[o]=aR[j]; p1[o]=aI[j];
  }
}

// Column DFT, complex<->complex, per image. SGN=+1: e^{-i t}, SGN=-1: e^{+i t}.
// Zre = C@Yre + s*S@Yim ; Zim = C@Yim - s*S@Yre
template<int SGN>
__global__ void __launch_bounds__(32) k_dft_cols(const float* inr, const float* ini,
                                                 const float* thc, const float* ths,
                                                 float* outr, float* outi)
{
  int lane = threadIdx.x, m = lane & 15, kh2 = lane >> 4;
  int img = blockIdx.x, khB = blockIdx.y*16, kxB = blockIdx.z*16;
  const float* yr = inr + (size_t)img*H_*WFP_;
  const float* yi = ini + (size_t)img*H_*WFP_;
  const float sg = (float)SGN;
  v8f aR = {}; v8f aI = {};
  for (int k0=0;k0<H_;k0+=4){
    int ka = k0 + 2*kh2;
    v2f ac = *(const v2f*)(thc + (khB+m)*H_ + ka);
    v2f as = *(const v2f*)(ths + (khB+m)*H_ + ka);
    v2f asp; asp.x = as.x*sg;  asp.y = as.y*sg;
    v2f asn; asn.x = -asp.x;   asn.y = -asp.y;
    v2f br, bi;
    br.x = yr[ka*WFP_ + kxB + m];  br.y = yr[(ka+1)*WFP_ + kxB + m];
    bi.x = yi[ka*WFP_ + kxB + m];  bi.y = yi[(ka+1)*WFP_ + kxB + m];
    aR = wmma4(ac,  br, aR);
    aR = wmma4(asp, bi, aR);
    aI = wmma4(ac,  bi, aI);
    aI = wmma4(asn, br, aI);
  }
  #pragma unroll
  for (int j=0;j<8;j++){
    size_t o = ((size_t)img*H_ + khB + j + 8*kh2)*WFP_ + kxB + m;
    outr[o]=aR[j]; outi[o]=aI[j];
  }
}

// Inverse row DFT (complex in, real out) with final scaling applied.
__global__ void __launch_bounds__(32) k_idft_rows(const float* zr, const float* zi,
                                                  const float* tiwc, const float* tiws,
                                                  const float* sc, float* out)
{
  int lane = threadIdx.x, m = lane & 15, kh2 = lane >> 4;
  int rowBase = blockIdx.x*16, wB = blockIdx.y*16;
  const float* zrr = zr + (size_t)rowBase*WFP_;
  const float* zir = zi + (size_t)rowBase*WFP_;
  v8f acc = {};
  for (int k0=0;k0<WFP_;k0+=4){
    int ka = k0 + 2*kh2;
    v2f aRe = *(const v2f*)(zrr + m*WFP_ + ka);
    v2f aIm = *(const v2f*)(zir + m*WFP_ + ka);
    v2f bc, bs;
    bc.x = tiwc[ka*W_ + wB + m];  bc.y = tiwc[(ka+1)*W_ + wB + m];
    bs.x = tiws[ka*W_ + wB + m];  bs.y = tiws[(ka+1)*W_ + wB + m];
    acc = wmma4(aRe, bc, acc);
    acc = wmma4(aIm, bs, acc);
  }
  int bt = rowBase / (C_*H_);                  // constant within a 16-row tile
  float s = sc[bt] * (1.f/16384.f);            // un-normalized fwd/inv -> /(H*W)
  #pragma unroll
  for (int j=0;j<8;j++)
    out[(size_t)(rowBase + j + 8*kh2)*W_ + wB + m] = acc[j]*s;
}

// --------------------------------------------------- projection / scan ------
__global__ void k_project(float* pr, float* pi, const float* kg)
{
  int i = blockIdx.x*blockDim.x + threadIdx.x;   // over BT*H*WFP
  if (i >= BT_*H_*WFP_) return;
  int hk = i % (H_*WFP_);
  int bt = i / (H_*WFP_);
  float gy=kg[3*hk], gx=kg[3*hk+1], k2=kg[3*hk+2];
  size_t b0 = (size_t)bt*C_*(H_*WFP_) + hk;
  size_t b1 = b0 + (size_t)(H_*WFP_);
  float ur=pr[b0], ui=pi[b0], vr=pr[b1], vi=pi[b1];
  float dr = (gy*ur + gx*vr)/k2;
  float di = (gy*ui + gx*vi)/k2;
  pr[b0]=ur-gy*dr; pi[b0]=ui-gy*di;
  pr[b1]=vr-gx*dr; pi[b1]=vi-gx*di;
}

// Serial scan over T per (b,h,k); projects output copy; accumulates |.|^2 norms.
__global__ void k_scan(const float* xr, const float* xi, const float* Ar, const float* Ai,
                       const float* kg, float* hr, float* hi, float* nin, float* nout)
{
  __shared__ float rin[256], rout[256];
  int tid = threadIdx.x;
  int gid = blockIdx.x*blockDim.x + tid;     // exactly B*H*WFP = 40960 threads
  int b  = gid / (H_*WFP_);
  int hk = gid % (H_*WFP_);
  float gy=kg[3*hk], gx=kg[3*hk+1], k2=kg[3*hk+2];
  float ar[C_], ai[C_], sr[C_], si[C_];
  #pragma unroll
  for (int c=0;c<C_;c++){
    size_t o = (size_t)(b*C_+c)*(H_*WFP_) + hk;
    ar[c]=Ar[o]; ai[c]=Ai[o]; sr[c]=0.f; si[c]=0.f;
  }
  for (int t=0;t<T_;t++){
    int bt = b*T_ + t;
    size_t base = (size_t)bt*C_*(H_*WFP_) + hk;
    if (t+1 < T_) {                            // prefetch next timestep's planes
      size_t pn = base + (size_t)C_*(H_*WFP_);
      __builtin_prefetch(xr + pn, 0, 1);
      __builtin_prefetch(xi + pn, 0, 1);
    }
    float lin=0.f, lout=0.f;
    float outr[C_], outi[C_];
    #pragma unroll
    for (int c=0;c<C_;c++){
      size_t o = base + (size_t)c*(H_*WFP_);
      float vr = xr[o], vi = xi[o];
      lin += vr*vr + vi*vi;
      float nr = ar[c]*sr[c] - ai[c]*si[c] + vr;
      float ni = ar[c]*si[c] + ai[c]*sr[c] + vi;
      sr[c]=nr; si[c]=ni; outr[c]=nr; outi[c]=ni;
    }
    float dr = (gy*outr[0] + gx*outr[1])/k2;
    float di = (gy*outi[0] + gx*outi[1])/k2;
    outr[0]-=gy*dr; outi[0]-=gy*di;
    outr[1]-=gx*dr; outi[1]-=gx*di;
    #pragma unroll
    for (int c=0;c<C_;c++){
      size_t o = base + (size_t)c*(H_*WFP_);
      hr[o]=outr[c]; hi[o]=outi[c];
      lout += outr[c]*outr[c] + outi[c]*outi[c];
    }
    rin[tid]=lin; rout[tid]=lout;
    __syncthreads();
    for (int s2=128;s2>0;s2>>=1){
      if (tid<s2){ rin[tid]+=rin[tid+s2]; rout[tid]+=rout[tid+s2]; }
      __syncthreads();
    }
    if (tid==0){ atomicAdd(&nin[bt], rin[0]); atomicAdd(&nout[bt], rout[0]); }
    __syncthreads();
  }
}

__global__ void k_scale(const float* nin, const float* nout, float* sc)
{
  int i = threadIdx.x;
  if (i < BT_){
    float a = sqrtf(nin[i])  * (1.f/128.f);   // convert to ortho norms (eps-exact)
    float b = sqrtf(nout[i]) * (1.f/128.f);
    sc[i] = a / (b + 1e-6f);
  }
}

// ----------------------------------------------------------------- launch ---
extern "C" void kernel_launch(void* const* d_in, const int* in_sizes, int n_in,
                              void* d_out, int out_size, void* d_ws, size_t ws_size,
                              hipStream_t stream)
{
  (void)in_sizes; (void)n_in; (void)out_size; (void)ws_size;
  const float* x_seq = (const float*)d_in[0];
  const float* dt    = (const float*)d_in[1];
  const float* pw1   = (const float*)d_in[2];
  const float* pb1   = (const float*)d_in[3];
  const float* pw2   = (const float*)d_in[4];
  const float* pb2   = (const float*)d_in[5];
  const float* w1    = (const float*)d_in[6];
  const float* b1    = (const float*)d_in[7];
  const float* w2    = (const float*)d_in[8];
  const float* b2    = (const float*)d_in[9];
  const float* w3    = (const float*)d_in[10];
  // d_in[11] (conv3_b) does not affect the gradient of a sum.

  float* Wp = (float*)d_ws;
  size_t o = 0;
  float* z1 = Wp + o; o += (size_t)BT_*C2_*HW_;   // 33.5M (later holds g1)
  float* z2 = Wp + o; o += (size_t)BT_*C2_*HW_;   // 33.5M (later holds g2)
  float* xt = Wp + o; o += (size_t)BT_*C_*HW_;    // 16.8M X_total
  float* p0 = Wp + o; o += NPLANE_;               // row-DFT re   -> later h_f re
  float* p1 = Wp + o; o += NPLANE_;               // row-DFT im   -> later h_f im
  float* p2 = Wp + o; o += NPLANE_;               // Xf re        -> later inv-col re
  float* p3 = Wp + o; o += NPLANE_;               // Xf im        -> later inv-col im
  float* Ar = Wp + o; o += (size_t)B_*C_*H_*WFP_;
  float* Ai = Wp + o; o += (size_t)B_*C_*H_*WFP_;
  float* twc = Wp + o; o += H_*WFP_;
  float* tws = Wp + o; o += H_*WFP_;
  float* thc = Wp + o; o += H_*H_;
  float* ths = Wp + o; o += H_*H_;
  float* tiwc = Wp + o; o += WFP_*W_;
  float* tiws = Wp + o; o += WFP_*W_;
  float* kg   = Wp + o; o += 3*H_*WFP_;
  float* nin  = Wp + o; o += BT_;
  float* nout = Wp + o; o += BT_;
  float* sc   = Wp + o; o += BT_;

  k_init<<<64,256,0,stream>>>(twc,tws,thc,ths,tiwc,tiws,kg,nin,nout);
  k_A<<<40,256,0,stream>>>(pw1,pb1,pw2,pb2,dt,kg,Ar,Ai);
  k_conv1<<<8192,256,0,stream>>>(x_seq,w1,b1,z1);
  k_conv2<<<8192,256,0,stream>>>(z1,w2,b2,z2);
  k_g2<<<131072,256,0,stream>>>(z2,w3);
  k_backT2<<<8192,256,0,stream>>>(z1,z2,w2);
  k_forceX<<<8192,256,0,stream>>>(x_seq,z1,w1,dt,xt);
  k_dft_rows<<<dim3(ROWS_/16,5),32,0,stream>>>(xt,twc,tws,p0,p1);
  k_dft_cols<1><<<dim3(BT_*C_,8,5),32,0,stream>>>(p0,p1,thc,ths,p2,p3);
  k_project<<<5120,256,0,stream>>>(p2,p3,kg);
  k_scan<<<160,256,0,stream>>>(p2,p3,Ar,Ai,kg,p0,p1,nin,nout);
  k_scale<<<1,128,0,stream>>>(nin,nout,sc);
  k_dft_cols<-1><<<dim3(BT_*C_,8,5),32,0,stream>>>(p0,p1,thc,ths,p2,p3);
  k_idft_rows<<<dim3(ROWS_/16,8),32,0,stream>>>(p2,p3,tiwc,tiws,sc,(float*)d_out);
}